// Embedder_78383153152126
// MI455X (gfx1250) — compile-verified
//
#include <hip/hip_runtime.h>

typedef __attribute__((ext_vector_type(2))) float v2f;
typedef __attribute__((ext_vector_type(8))) float v8f;

#define B_DIM     4096
#define STATE_DIM 512
#define N_NODES   64
#define OB        32
#define EMBED     128

// One wave per (16-row batch chunk, node). Computes a 16x128 f32 output strip
// via 8 column tiles x 8 K-steps of V_WMMA_F32_16X16X4_F32.
__global__ __launch_bounds__(32) void embedder_wmma(
    const float* __restrict__ state,       // [4096, 512]
    const int*   __restrict__ gather_idx,  // [64, 32]
    const int*   __restrict__ node_type,   // [64]
    const float* __restrict__ W,           // [4, 32, 128]
    const float* __restrict__ bias,        // [4, 128]
    float*       __restrict__ out)         // [4096, 64, 128]
{
  const int lane   = threadIdx.x;              // 0..31, wave32
  const int node   = blockIdx.x & (N_NODES - 1);
  const int mchunk = blockIdx.x >> 6;          // 0..255

  const int half  = lane >> 4;                 // 0 = lanes 0-15, 1 = lanes 16-31
  const int nlo   = lane & 15;                 // A row / B col / C col within tile
  const int khalf = half * 2;                  // K sub-offset per ISA A/B layout

  const int type = node_type[node];
  const float* Wt   = W + type * (OB * EMBED);
  const float* srow = state + (size_t)(mchunk * 16 + nlo) * STATE_DIM;

  // ---- A fragments: lane l holds (m = l&15, k = 4*kk + 2*(l>>4) + v) ----
  v2f a[8];
  const int* gi = gather_idx + node * OB;
  #pragma unroll
  for (int kk = 0; kk < 8; ++kk) {
    const int ka = kk * 4 + khalf;
    const int ia = gi[ka];
    const int ib = gi[ka + 1];
    a[kk].x = srow[ia];
    a[kk].y = srow[ib];
  }

  const int mhi = half * 8;                    // C/D row offset for this half-wave

  #pragma unroll
  for (int nt = 0; nt < 8; ++nt) {
    const int e  = nt * 16 + nlo;              // embed column
    const float bv = bias[type * EMBED + e];
    // C(m, e) = b[type][e] for all m -> every acc VGPR starts at bv
    v8f acc = {bv, bv, bv, bv, bv, bv, bv, bv};

    #pragma unroll
    for (int kk = 0; kk < 8; ++kk) {
      const int kb = kk * 4 + khalf;           // B: lane = n + 16*(k>>1), vgpr = k&1
      v2f bf;
      bf.x = Wt[kb * EMBED + e];
      bf.y = Wt[(kb + 1) * EMBED + e];
      acc = __builtin_amdgcn_wmma_f32_16x16x4_f32(
          /*neg_a=*/false, a[kk], /*neg_b=*/false, bf,
          /*c_mod=*/(short)0, acc, /*reuse_a=*/false, /*reuse_b=*/false);
    }

    // D layout: VGPR r -> row m = r + 8*half, col = e (lane&15)
    #pragma unroll
    for (int r = 0; r < 8; ++r) {
      const int m = mchunk * 16 + mhi + r;
      out[(size_t)m * (N_NODES * EMBED) + node * EMBED + e] = acc[r];
    }
  }
}

extern "C" void kernel_launch(void* const* d_in, const int* in_sizes, int n_in,
                              void* d_out, int out_size, void* d_ws, size_t ws_size,
                              hipStream_t stream) {
  const float* state = (const float*)d_in[0];
  const int*   gidx  = (const int*)d_in[1];
  const int*   ntype = (const int*)d_in[2];
  const float* W     = (const float*)d_in[3];
  const float* b     = (const float*)d_in[4];
  float*       out   = (float*)d_out;

  dim3 grid((B_DIM / 16) * N_NODES);   // 16384 single-wave workgroups
  dim3 block(32);
  embedder_wmma<<<grid, block, 0, stream>>>(state, gidx, ntype, W, b, out);
}